// HandshakingKernel_37752762532348
// MI455X (gfx1250) — compile-verified
//
#include <hip/hip_runtime.h>
#include <hip/hip_bf16.h>
#include <cstdint>

#define B_  2
#define S_  256
#define H_  512
#define P_  ((S_*(S_+1))/2)       // 32896
#define NEGF (-1e30f)
#define EPSF (1e-12f)
#define RS  520                    // padded LDS row stride (floats): avoids 16-way bank conflicts, keeps 8B align

typedef __attribute__((ext_vector_type(2))) float v2f;
typedef __attribute__((ext_vector_type(8))) float v8f;

// ---- workspace layout (float offsets), total 8 MB ----
#define OFF_SEQM  0
#define OFF_CSUM  (OFF_SEQM + B_*S_*H_)
#define OFF_WG1P  (OFF_CSUM + B_*S_*H_)
#define OFF_WB1P  (OFF_WG1P + H_*H_)
#define OFF_WG2P  (OFF_WB1P + H_*H_)
#define OFF_WB2P  (OFF_WG2P + H_*H_)
#define OFF_G1    (OFF_WB2P + H_*H_)
#define OFF_B1    (OFF_G1   + B_*S_*H_)

// ---------------- K0: masked seq + inclusive cumsum over S ----------------
__global__ void k_prep(const float* __restrict__ x, const float* __restrict__ mask,
                       float* __restrict__ seqm, float* __restrict__ csum) {
  int b = blockIdx.y;
  int h = blockIdx.x * blockDim.x + threadIdx.x;
  float acc = 0.f;
  for (int s = 0; s < S_; ++s) {
    size_t idx = ((size_t)(b * S_ + s)) * H_ + h;
    float v = x[idx] + (1.0f - mask[b * S_ + s]) * (-1000.0f);
    seqm[idx] = v;
    acc += v;
    csum[idx] = acc;
  }
}

// ---------------- K1: repack H x H weights into WMMA B-fragment order ----------------
// Wp[((kt*32 + ntile)*32 + lane)*2 + c] = W[(4*kt + 2*(lane>>4) + c)][16*ntile + (lane&15)]
__global__ void k_repack(const float* __restrict__ Wg1, const float* __restrict__ Wb1,
                         const float* __restrict__ Wg2, const float* __restrict__ Wb2,
                         float* __restrict__ ws) {
  const float* src; float* dst;
  switch (blockIdx.y) {
    case 0:  src = Wg1; dst = ws + OFF_WG1P; break;
    case 1:  src = Wb1; dst = ws + OFF_WB1P; break;
    case 2:  src = Wg2; dst = ws + OFF_WG2P; break;
    default: src = Wb2; dst = ws + OFF_WB2P; break;
  }
  int t    = blockIdx.x * blockDim.x + threadIdx.x;  // 0 .. 131071
  int lane = t & 31;
  int n    = (t >> 5) & 31;
  int kt   = t >> 10;
  int hi   = lane >> 4;
  int col  = (n << 4) + (lane & 15);
  int k0   = (kt << 2) + (hi << 1);
  float2 v;
  v.x = src[(size_t)k0 * H_ + col];
  v.y = src[(size_t)(k0 + 1) * H_ + col];
  *(float2*)(dst + (size_t)t * 2) = v;
}

// ---------------- K2: G1/B1 = bias + seqm @ Wg1/Wb1 (f32 WMMA) ----------------
__global__ void __launch_bounds__(256) k_gemm_rows(
    const float* __restrict__ seqm,
    const float* __restrict__ Wg1p, const float* __restrict__ Wb1p,
    const float* __restrict__ g1,   const float* __restrict__ b1,
    float* __restrict__ G1, float* __restrict__ B1) {
  __shared__ __attribute__((aligned(16))) float sA[16 * RS];
  int tid = threadIdx.x;
  int R0  = blockIdx.x * 16;
  for (int t = tid; t < 16 * H_; t += 256) {
    int r = t >> 9, c = t & (H_ - 1);
    sA[r * RS + c] = seqm[(size_t)(R0 + r) * H_ + c];
  }
  __syncthreads();

  int wave = tid >> 5, lane = tid & 31, hi = lane >> 4;
  v8f zero = {0.f,0.f,0.f,0.f,0.f,0.f,0.f,0.f};
  v8f accG[4] = {zero, zero, zero, zero};
  v8f accB[4] = {zero, zero, zero, zero};

  for (int kt = 0; kt < 128; ++kt) {
    v2f a = *(const v2f*)(&sA[(lane & 15) * RS + (kt << 2) + (hi << 1)]);
    #pragma unroll
    for (int n = 0; n < 4; ++n) {
      int fo = ((((kt << 5) + (wave << 2) + n) << 5) | lane) << 1;
      v2f bg = *(const v2f*)(Wg1p + fo);
      v2f bb = *(const v2f*)(Wb1p + fo);
      accG[n] = __builtin_amdgcn_wmma_f32_16x16x4_f32(false, a, false, bg, (short)0, accG[n], false, false);
      accB[n] = __builtin_amdgcn_wmma_f32_16x16x4_f32(false, a, false, bb, (short)0, accB[n], false, false);
    }
  }
  #pragma unroll
  for (int n = 0; n < 4; ++n) {
    int col = (((wave << 2) + n) << 4) + (lane & 15);
    float gbias = g1[col], bbias = b1[col];
    #pragma unroll
    for (int v = 0; v < 8; ++v) {
      int M = v + (hi << 3);
      size_t o = (size_t)(R0 + M) * H_ + col;
      G1[o] = gbias + accG[n][v];
      B1[o] = bbias + accB[n][v];
    }
  }
}

// ---------------- K3: fused inner + condLN1 + WMMA GEMM + condLN2 ----------------
__global__ void __launch_bounds__(256) k_main(
    const float* __restrict__ seqm, const float* __restrict__ csum,
    const float* __restrict__ G1,   const float* __restrict__ B1,
    const float* __restrict__ Wg2p, const float* __restrict__ Wb2p,
    const float* __restrict__ lamtha,
    const float* __restrict__ g2,   const float* __restrict__ b2,
    float* __restrict__ out) {
  int jt = blockIdx.x, i = blockIdx.y, b = blockIdx.z;
  int j0 = jt << 4;
  if (j0 + 15 < i) return;   // block-uniform: EXEC stays all-ones for WMMA

  __shared__ __attribute__((aligned(16))) float sInner[16 * RS];
  __shared__ __attribute__((aligned(16))) float sShak[16 * RS];
  __shared__ float sG1[H_], sB1[H_];
  __shared__ float sMu2[16], sInv2[16];

  int tid = threadIdx.x;
  const float* seqb = seqm + (size_t)b * S_ * H_;
  const float* csb  = csum + (size_t)b * S_ * H_;

  // ---- Phase A: inner = lam*mean + (1-lam)*max over window [i..j], j in tile ----
  {
    int jstart = (i > j0) ? i : j0;
    int c0 = tid, c1 = tid + 256;
    float lam0 = lamtha[c0], lam1 = lamtha[c1];
    float si0 = seqb[(size_t)i * H_ + c0], si1 = seqb[(size_t)i * H_ + c1];
    float ci0 = csb[(size_t)i * H_ + c0],  ci1 = csb[(size_t)i * H_ + c1];
    float m0 = NEGF, m1 = NEGF;
    for (int s = i; s < jstart; ++s) {
      m0 = fmaxf(m0, seqb[(size_t)s * H_ + c0]);
      m1 = fmaxf(m1, seqb[(size_t)s * H_ + c1]);
    }
    for (int r = 0; r < 16; ++r) {
      int j = j0 + r;
      float in0 = 0.f, in1 = 0.f;
      if (j >= i) {
        m0 = fmaxf(m0, seqb[(size_t)j * H_ + c0]);
        m1 = fmaxf(m1, seqb[(size_t)j * H_ + c1]);
        float len = (float)(j - i + 1);
        float mean0 = (csb[(size_t)j * H_ + c0] - ci0 + si0) / len;
        float mean1 = (csb[(size_t)j * H_ + c1] - ci1 + si1) / len;
        in0 = lam0 * mean0 + (1.f - lam0) * m0;
        in1 = lam1 * mean1 + (1.f - lam1) * m1;
      }
      sInner[r * RS + c0] = in0;
      sInner[r * RS + c1] = in1;
    }
    size_t gro = (size_t)(b * S_ + i) * H_;
    sG1[c0] = G1[gro + c0];  sG1[c1] = G1[gro + c1];
    sB1[c0] = B1[gro + c0];  sB1[c1] = B1[gro + c1];
  }
  __syncthreads();

  // ---- Phase B: shaking = condLN(visible; G1row,B1row); stats of shaking ----
  int wave = tid >> 5, lane = tid & 31;
  for (int rr = 0; rr < 2; ++rr) {
    int r = (wave << 1) + rr;
    int j = j0 + r;
    bool valid = (j >= i);
    float vals[16];
    float sum = 0.f, ssq = 0.f;
    #pragma unroll
    for (int q = 0; q < 16; ++q) {
      int c = lane + (q << 5);
      float v = valid ? seqb[(size_t)j * H_ + c] : 0.f;
      vals[q] = v; sum += v; ssq += v * v;
    }
    #pragma unroll
    for (int o = 16; o > 0; o >>= 1) { sum += __shfl_xor(sum, o, 32); ssq += __shfl_xor(ssq, o, 32); }
    float mu  = sum * (1.f / H_);
    float inv = rsqrtf(fmaxf(ssq * (1.f / H_) - mu * mu, 0.f) + EPSF);
    float sum2 = 0.f, ssq2 = 0.f;
    #pragma unroll
    for (int q = 0; q < 16; ++q) {
      int c = lane + (q << 5);
      float sh = (vals[q] - mu) * inv * sG1[c] + sB1[c];
      sShak[r * RS + c] = sh;
      sum2 += sh; ssq2 += sh * sh;
    }
    #pragma unroll
    for (int o = 16; o > 0; o >>= 1) { sum2 += __shfl_xor(sum2, o, 32); ssq2 += __shfl_xor(ssq2, o, 32); }
    if (lane == 0) {
      float mu2 = sum2 * (1.f / H_);
      sMu2[r]  = mu2;
      sInv2[r] = rsqrtf(fmaxf(ssq2 * (1.f / H_) - mu2 * mu2, 0.f) + EPSF);
    }
  }
  __syncthreads();

  // ---- Phase C: G2/B2 accumulators = inner @ Wg2 / inner @ Wb2 (f32 WMMA) ----
  int hi = lane >> 4;
  v8f zero = {0.f,0.f,0.f,0.f,0.f,0.f,0.f,0.f};
  v8f accG[4] = {zero, zero, zero, zero};
  v8f accB[4] = {zero, zero, zero, zero};

  for (int kt = 0; kt < 128; ++kt) {
    v2f a = *(const v2f*)(&sInner[(lane & 15) * RS + (kt << 2) + (hi << 1)]);
    #pragma unroll
    for (int n = 0; n < 4; ++n) {
      int fo = ((((kt << 5) + (wave << 2) + n) << 5) | lane) << 1;
      v2f bg = *(const v2f*)(Wg2p + fo);
      v2f bb = *(const v2f*)(Wb2p + fo);
      accG[n] = __builtin_amdgcn_wmma_f32_16x16x4_f32(false, a, false, bg, (short)0, accG[n], false, false);
      accB[n] = __builtin_amdgcn_wmma_f32_16x16x4_f32(false, a, false, bb, (short)0, accB[n], false, false);
    }
  }

  // ---- Phase D: fused second condLN epilogue + store ----
  int p_base = i * S_ - (i * (i - 1)) / 2;
  #pragma unroll
  for (int n = 0; n < 4; ++n) {
    int col = (((wave << 2) + n) << 4) + (lane & 15);
    float gg = g2[col], bbias = b2[col];
    #pragma unroll
    for (int v = 0; v < 8; ++v) {
      int M = v + (hi << 3);
      int j = j0 + M;
      if (j >= i) {
        int p = p_base + (j - i);
        float sh = sShak[M * RS + col];
        float val = (sh - sMu2[M]) * sInv2[M] * (gg + accG[n][v]) + (bbias + accB[n][v]);
        out[((size_t)b * P_ + p) * H_ + col] = val;
      }
    }
  }
}

extern "C" void kernel_launch(void* const* d_in, const int* in_sizes, int n_in,
                              void* d_out, int out_size, void* d_ws, size_t ws_size,
                              hipStream_t stream) {
  (void)in_sizes; (void)n_in; (void)out_size; (void)ws_size;
  const float* x    = (const float*)d_in[0];
  const float* mask = (const float*)d_in[1];
  const float* lam  = (const float*)d_in[2];
  const float* Wg1  = (const float*)d_in[3];
  const float* Wb1  = (const float*)d_in[4];
  const float* g1   = (const float*)d_in[5];
  const float* b1   = (const float*)d_in[6];
  const float* Wg2  = (const float*)d_in[7];
  const float* Wb2  = (const float*)d_in[8];
  const float* g2   = (const float*)d_in[9];
  const float* b2   = (const float*)d_in[10];
  float* ws  = (float*)d_ws;
  float* out = (float*)d_out;

  k_prep<<<dim3(H_ / 256, B_), 256, 0, stream>>>(x, mask, ws + OFF_SEQM, ws + OFF_CSUM);
  k_repack<<<dim3((H_ * H_ / 2) / 256, 4), 256, 0, stream>>>(Wg1, Wb1, Wg2, Wb2, ws);
  k_gemm_rows<<<dim3((B_ * S_) / 16), 256, 0, stream>>>(
      ws + OFF_SEQM, ws + OFF_WG1P, ws + OFF_WB1P, g1, b1, ws + OFF_G1, ws + OFF_B1);
  k_main<<<dim3(S_ / 16, S_, B_), 256, 0, stream>>>(
      ws + OFF_SEQM, ws + OFF_CSUM, ws + OFF_G1, ws + OFF_B1,
      ws + OFF_WG2P, ws + OFF_WB2P, lam, g2, b2, out);
}